// HMM_21500606284162
// MI455X (gfx1250) — compile-verified
//
#include <hip/hip_runtime.h>
#include <math.h>

typedef float v2f __attribute__((ext_vector_type(2)));
typedef float v8f __attribute__((ext_vector_type(8)));

#define N_EFF_C   8570000.0f
#define S_C       41285.0f
#define SIGMA_C   2.5f
#define TC_C      22.0f
#define TR_C      100.0f

// ---------------------------------------------------------------------------
// Phase 1: sequential nonlinear scan. One wave, lane 0 carries the 8-float
// state entirely in VGPRs. Per step: 2 independent v_exp_f32 (the
// base**(e*x2)*base**(e*x3) term), one sqrt for CH=(x0+x4)^2.5 (s^2*sqrt(s)),
// ~14 FMAs for the sparse 7x7 matvec, and two fire-and-forget
// global_store_b128 of the state row. Latency-bound on the serial chain;
// nothing else on the chip can help a non-associative scan.
// ---------------------------------------------------------------------------
__global__ __launch_bounds__(32)
void hmm_scan_kernel(const float* __restrict__ prior,
                     const float* __restrict__ eta_p,
                     const float* __restrict__ alpha_p,
                     const float* __restrict__ mu_p,
                     const float* __restrict__ gamma_p,
                     const float* __restrict__ phi_p,
                     const float* __restrict__ w_p,
                     const float* __restrict__ xi_p,
                     const float* __restrict__ beta_p,
                     const float* __restrict__ k_p,
                     const float* __restrict__ C_p,
                     const float* __restrict__ eps_p,
                     const float* __restrict__ kappa_p,
                     float* __restrict__ hs, int T)
{
    if (threadIdx.x != 0 || blockIdx.x != 0) return;

    const float eta   = *eta_p;
    const float a     = alpha_p[0];
    const float mu    = *mu_p;
    const float gam   = *gamma_p;
    const float phi   = *phi_p;
    const float w     = *w_p;
    const float xi    = *xi_p;
    const float beta  = *beta_p;
    const float k     = *k_p;
    const float C     = *C_p;
    const float eps   = *eps_p;
    const float kap0  = kappa_p[0];

    // f_ratio = 2 - exp(-eps * N_eff/S); z = 1/f_ratio; e = z*k_eff*f_ratio*C
    const float f_ratio = 2.0f - expf(-eps * (N_EFF_C / S_C));
    const float z       = 1.0f / f_ratio;
    const float e_k     = ((z * k) * f_ratio) * C;                          // t<TC or t>=TR
    const float e_kp    = ((z * (kap0 * (SIGMA_C - 1.0f))) * f_ratio) * C;  // TC<=t<TR
    const float lb      = log2f(1.0f - beta);   // log2 of pow base

    // loop-invariant transition coefficients
    const float c11 = 1.0f - eta;
    const float c22 = 1.0f - a;
    const float c33 = 1.0f - mu;
    const float c43 = mu * gam;
    const float c44 = w * (1.0f - phi) + (1.0f - w) * (1.0f - xi);
    const float c53 = mu * (1.0f - gam);
    const float c54 = (1.0f - w) * xi;
    const float c64 = w * phi;

    float x0 = prior[0], x1 = prior[1], x2 = prior[2], x3 = prior[3];
    float x4 = 0.0f, x5 = 0.0f, x6 = 0.0f;
    float t  = 1.0f;

    for (int s = 0; s < T; ++s) {
        // record hidden state (pre-step), 8 floats, two 16B stores
        float4* hrow = (float4*)(hs + (size_t)s * 8);
        hrow[0] = make_float4(x0, x1, x2, x3);
        hrow[1] = make_float4(x4, x5, x6, t);

        const bool  mid = (t >= TC_C) && (t < TR_C);
        const float e   = mid ? e_kp : e_k;
        // Gamma = 1 - base^(e*x2) * base^(e*x3)
        const float g1    = exp2f(lb * (e * x2));
        const float g2    = exp2f(lb * (e * x3));
        const float Gamma = 1.0f - g1 * g2;

        const float sh = x0 + x4;
        const float CH = (sh * sh) * sqrtf(sh);        // sh^2.5
        const float oneMG = 1.0f - Gamma;
        const float a00 = (t == TC_C) ? (1.0f - CH) * oneMG : oneMG;

        const float nx0 = a00 * x0;
        const float nx1 = (1.0f - a00) * x0 + c11 * x1;
        const float nx2 = eta * x1 + c22 * x2;
        const float nx3 = a   * x2 + c33 * x3;
        const float nx4 = c43 * x3 + c44 * x4;
        const float nx5 = c53 * x3 + c54 * x4 + x5;
        const float nx6 = c64 * x4 + x6;

        x0 = nx0; x1 = nx1; x2 = nx2; x3 = nx3;
        x4 = nx4; x5 = nx5; x6 = nx6;
        t += 1.0f;
    }
}

// ---------------------------------------------------------------------------
// Phase 2: Y[T,3] = H[T,7] @ obs^T[7,3] via V_WMMA_F32_16X16X4_F32.
// One wave per 16-row tile of H; K=7 split into two K=4 chunks (zero padded).
// A layout (32-bit 16x4): lane<16 -> (M=lane, K={0,1}); lane>=16 -> K={2,3}.
// B layout mirrors A over K rows, N = lane&15. D: VGPR r holds rows r / r+8.
// All operand selection is branchless (v_cndmask), so EXEC stays all-ones
// from kernel entry through both WMMAs; the only lane-divergent region is
// the final store block.
// ---------------------------------------------------------------------------
__global__ __launch_bounds__(256)
void hmm_ys_wmma_kernel(const float* __restrict__ hs,
                        const float* __restrict__ obs,   // 3x7 row-major
                        float* __restrict__ ys, int T)
{
    const int lane = threadIdx.x & 31;
    const int wave = threadIdx.x >> 5;
    const int tile = blockIdx.x * (blockDim.x >> 5) + wave;
    const int row0 = tile * 16;
    if (row0 >= T) return;                 // wave-uniform: EXEC stays all-ones

    const bool hi_half = (lane & 16) != 0; // lanes 16-31
    const int  m       = lane & 15;        // row-in-tile (A) / N column (B,D)

    int rm = row0 + m;
    if (rm > T - 1) rm = T - 1;            // safe load clamp (stores guarded)

    // ---- A: two b128 loads of the 8-float hs row, branchless half select
    const float4* hrow4 = (const float4*)(hs + (size_t)rm * 8);
    const float4 lo = hrow4[0];            // x0..x3
    const float4 hi = hrow4[1];            // x4,x5,x6,t
    v2f a0, a1;
    a0.x = hi_half ? lo.z : lo.x;          // K=2 : K=0
    a0.y = hi_half ? lo.w : lo.y;          // K=3 : K=1
    a1.x = hi_half ? hi.z : hi.x;          // K=6 : K=4
    a1.y = hi_half ? 0.0f : hi.y;          // pad : K=5   (never read col 7)

    // ---- B: B[k][n] = obs[n][k] for n<3 else 0; clamp n so loads are
    // always in-bounds, then mask with selects (no EXEC manipulation).
    const int  n  = m;
    const bool nv = (n < 3);
    const float* orow = obs + (nv ? n : 0) * 7;
    const int  k0 = hi_half ? 2 : 0;
    const float v00 = orow[k0];            // K=k0
    const float v01 = orow[k0 + 1];        // K=k0+1
    const float v10 = orow[k0 + 4];        // K=k0+4 (4 or 6)
    const float v11 = orow[5];             // K=5 (only low half uses it)
    v2f b0, b1;
    b0.x = nv ? v00 : 0.0f;
    b0.y = nv ? v01 : 0.0f;
    b1.x = nv ? v10 : 0.0f;
    b1.y = (nv && !hi_half) ? v11 : 0.0f;  // high half K=7 -> pad

    v8f acc = {0.0f, 0.0f, 0.0f, 0.0f, 0.0f, 0.0f, 0.0f, 0.0f};
    acc = __builtin_amdgcn_wmma_f32_16x16x4_f32(
        false, a0, false, b0, (short)0, acc, false, false);
    acc = __builtin_amdgcn_wmma_f32_16x16x4_f32(
        false, a1, false, b1, (short)0, acc, false, false);

    // ---- Store D: VGPR r holds rows (row0 + half*8 + r), column n.
    const int rbase = row0 + (hi_half ? 8 : 0);
    if (row0 + 16 <= T) {                  // wave-uniform fast path: no guards
        if (nv) {
            float* col = ys + (size_t)rbase * 3 + n;
            #pragma unroll
            for (int r = 0; r < 8; ++r) col[r * 3] = acc[r];
        }
    } else {                               // ragged tail tile
        if (nv) {
            #pragma unroll
            for (int r = 0; r < 8; ++r) {
                const int row = rbase + r;
                if (row < T) ys[(size_t)row * 3 + n] = acc[r];
            }
        }
    }
}

// ---------------------------------------------------------------------------
// d_in order: 0=T(int),1=prior,2=eta,3=alpha,4=mu,5=gamma,6=phi,7=w,8=xi,
//             9=beta,10=k,11=C,12=eps,13=kappa,14=obs
// d_out: [ys (T*3) | hs (T*8)]  => out_size = 11*T
// ---------------------------------------------------------------------------
extern "C" void kernel_launch(void* const* d_in, const int* in_sizes, int n_in,
                              void* d_out, int out_size, void* d_ws, size_t ws_size,
                              hipStream_t stream)
{
    (void)in_sizes; (void)n_in; (void)d_ws; (void)ws_size;

    const float* prior = (const float*)d_in[1];
    const float* eta   = (const float*)d_in[2];
    const float* alpha = (const float*)d_in[3];
    const float* mu    = (const float*)d_in[4];
    const float* gamma = (const float*)d_in[5];
    const float* phi   = (const float*)d_in[6];
    const float* w     = (const float*)d_in[7];
    const float* xi    = (const float*)d_in[8];
    const float* beta  = (const float*)d_in[9];
    const float* k     = (const float*)d_in[10];
    const float* C     = (const float*)d_in[11];
    const float* eps   = (const float*)d_in[12];
    const float* kappa = (const float*)d_in[13];
    const float* obs   = (const float*)d_in[14];

    const int T = out_size / 11;           // ys: 3*T, hs: 8*T
    float* ys = (float*)d_out;
    float* hs = (float*)d_out + (size_t)T * 3;

    hmm_scan_kernel<<<1, 32, 0, stream>>>(prior, eta, alpha, mu, gamma, phi,
                                          w, xi, beta, k, C, eps, kappa, hs, T);

    const int tiles  = (T + 15) / 16;
    const int wpb    = 8;                  // waves per block (256 threads)
    const int blocks = (tiles + wpb - 1) / wpb;
    hmm_ys_wmma_kernel<<<blocks, wpb * 32, 0, stream>>>(hs, obs, ys, T);
}